// MathAdaptivePositionalEncoding_22359599743549
// MI455X (gfx1250) — compile-verified
//
#include <hip/hip_runtime.h>
#include <math.h>

#define SEQ   4096
#define NB    4
#define DIM   1024
#define NPAIR 512   // DIM/2

typedef float __attribute__((ext_vector_type(4))) vfloat4;  // native vector for NT store

// ---------------------------------------------------------------------------
// Kernel 1: softmax of scale_weights (5) and comb_weights (5) -> wts[10]
// wts[0..4] = softmax(scale_weights), wts[5..9] = softmax(comb_weights)
// ---------------------------------------------------------------------------
__global__ void softmax2_kernel(const float* __restrict__ sw,
                                const float* __restrict__ cw,
                                float* __restrict__ wout) {
    if (threadIdx.x == 0 && blockIdx.x == 0) {
        float m = -3.4e38f;
        for (int i = 0; i < 5; ++i) m = fmaxf(m, sw[i]);
        float e[5]; float sum = 0.f;
        for (int i = 0; i < 5; ++i) { e[i] = expf(sw[i] - m); sum += e[i]; }
        for (int i = 0; i < 5; ++i) wout[i] = e[i] / sum;
        m = -3.4e38f;
        for (int i = 0; i < 5; ++i) m = fmaxf(m, cw[i]);
        sum = 0.f;
        for (int i = 0; i < 5; ++i) { e[i] = expf(cw[i] - m); sum += e[i]; }
        for (int i = 0; i < 5; ++i) wout[5 + i] = e[i] / sum;
    }
}

// ---------------------------------------------------------------------------
// Kernel 2: clamped bracket-depth prefix scan per batch row.
// Each token is a transfer function f(x)=clamp(x+d,0,3); these compose:
//   (a2,l2,h2) o (a1,l1,h1) = (a1+a2, clamp(l1+a2,l2,h2), clamp(h1+a2,l2,h2))
// 256 threads x 16 tokens, Hillis-Steele scan of transfer functions in LDS.
// ---------------------------------------------------------------------------
__global__ void level_scan_kernel(const int* __restrict__ tok,
                                  int* __restrict__ levels) {
    const int b   = blockIdx.x;
    const int tid = threadIdx.x;
    const int* t  = tok + b * SEQ + tid * 16;

    int toks[16];
    int a = 0, l = 0, h = 3;
#pragma unroll
    for (int i = 0; i < 16; ++i) {
        int v = t[i];
        toks[i] = v;
        int d = (v == 40 || v == 91 || v == 123) ? 1
              : ((v == 41 || v == 93 || v == 125) ? -1 : 0);
        a += d;
        l = min(max(l + d, 0), 3);
        h = min(max(h + d, 0), 3);
    }

    __shared__ int sa[256], sl[256], sh[256];
    sa[tid] = a; sl[tid] = l; sh[tid] = h;
    __syncthreads();

    int ca = a, cl = l, ch = h;
    for (int off = 1; off < 256; off <<= 1) {
        int pa = 0, pl = 0, ph = 3;
        const bool has = (tid >= off);
        if (has) { pa = sa[tid - off]; pl = sl[tid - off]; ph = sh[tid - off]; }
        __syncthreads();
        if (has) {
            int na = pa + ca;
            int nl = min(max(pl + ca, cl), ch);
            int nh = min(max(ph + ca, cl), ch);
            ca = na; cl = nl; ch = nh;
            sa[tid] = ca; sl[tid] = cl; sh[tid] = ch;
        }
        __syncthreads();
    }

    // exclusive prefix applied to the initial level 0: clamp(A, L, H)
    int lvl = (tid == 0) ? 0 : min(max(sa[tid - 1], sl[tid - 1]), sh[tid - 1]);

    int* lo = levels + b * SEQ + tid * 16;
#pragma unroll
    for (int i = 0; i < 16; ++i) {
        int v = toks[i];
        if (v == 40 || v == 91 || v == 123)      lvl = min(lvl + 1, 3);
        else if (v == 41 || v == 93 || v == 125) lvl = max(lvl - 1, 0);
        lo[i] = lvl;
    }
}

// ---------------------------------------------------------------------------
// Kernel 3: batch-independent sinusoid plane msad[S][D]:
//   msad[s,2p]   = cw0*sin(pos*freq_p) + cw3*0.1*sum_k w_k sin(pos/scale_k*div_p)
//   msad[s,2p+1] = same with cos
// div_p = 10000^(-p/512); freq_p = div_p * 1.25 * freq_adjust[p]
// ---------------------------------------------------------------------------
__global__ void msad_kernel(const float* __restrict__ freq_adjust,
                            const float* __restrict__ wts,
                            float2* __restrict__ msad) {
    const int idx = blockIdx.x * blockDim.x + threadIdx.x;  // = s*512 + p
    const int p   = idx & (NPAIR - 1);
    const int s   = idx >> 9;
    const float pos = (float)s;

    const float w0 = wts[0], w1 = wts[1], w2 = wts[2], w3 = wts[3], w4 = wts[4];
    const float cw0 = wts[5], cw3s = wts[8] * 0.1f;

    const float dv = expf((float)p * (-9.210340371976184f / 512.0f));

    float mss = 0.f, msc = 0.f, sn, cs;
    sincosf(pos * dv,            &sn, &cs); mss += w0 * sn; msc += w0 * cs;
    sincosf(pos * 0.5f    * dv,  &sn, &cs); mss += w1 * sn; msc += w1 * cs;
    sincosf(pos * 0.25f   * dv,  &sn, &cs); mss += w2 * sn; msc += w2 * cs;
    sincosf(pos * 0.125f  * dv,  &sn, &cs); mss += w3 * sn; msc += w3 * cs;
    sincosf(pos * 0.0625f * dv,  &sn, &cs); mss += w4 * sn; msc += w4 * cs;

    const float fr = dv * 1.25f * freq_adjust[p];
    float asn, acs;
    sincosf(pos * fr, &asn, &acs);

    float2 o;
    o.x = cw0 * asn + cw3s * mss;
    o.y = cw0 * acs + cw3s * msc;
    msad[idx] = o;
}

// ---------------------------------------------------------------------------
// Kernel 4: fused combine. One block per s; the msad row (4KB) is staged
// global->LDS with the gfx1250 ASYNC path (ASYNCcnt) and reused for all 4
// batches. Token/level loads + global_prefetch of the symbol rows are issued
// while the async copy is in flight. Output uses non-temporal b128 stores
// (write-once data, keep L2 for the reused msad plane / tables).
//   out[b,s,:] = msad[s,:] + cw4*sym[tok] + cw2*0.15*lvl_emb[level]
//              + opmask * cw1*0.2*prec_emb[pid]
// ---------------------------------------------------------------------------
__global__ void combine_kernel(const int* __restrict__ tok,
                               const float* __restrict__ symtab,
                               const float* __restrict__ lvltab,
                               const float* __restrict__ prectab,
                               const float* __restrict__ wts,
                               const float* __restrict__ msad,
                               const int* __restrict__ levels,
                               float* __restrict__ out) {
    const int s   = blockIdx.x;
    const int tid = threadIdx.x;

    __shared__ float lms[DIM];

    // --- CDNA5 async global->LDS copy of msad[s, :] (b128 per lane) ---
    {
        unsigned ldsoff = (unsigned)(size_t)(&lms[tid * 4]);
        unsigned long long gaddr =
            (unsigned long long)(size_t)(msad + (size_t)s * DIM + tid * 4);
        asm volatile("global_load_async_to_lds_b128 %0, %1, off"
                     :: "v"(ldsoff), "v"(gaddr) : "memory");
    }

    // Overlap: gather per-batch scalars + prefetch symbol rows while the
    // async copy is outstanding.
    int  t[NB], lvl[NB], pid[NB];
    bool opm[NB];
#pragma unroll
    for (int b = 0; b < NB; ++b) {
        const int tv = tok[b * SEQ + s];
        t[b]   = tv;
        lvl[b] = levels[b * SEQ + s];
        opm[b] = ((tv >= 40) && (tv <= 47)) || ((tv >= 60) && (tv <= 62)) || (tv == 94);
        int p = 0;
        if      (tv == 42 || tv == 47) p = 5;
        else if (tv == 94)             p = 6;
        else if (tv == 43 || tv == 45) p = 3;
        else if (tv == 60 || tv == 62) p = 2;
        else if (tv == 61)             p = 1;
        pid[b] = p;
        __builtin_prefetch(symtab + (size_t)tv * DIM + tid * 4, 0, 3);
    }

    const float cw1s = wts[6] * 0.2f;
    const float cw2s = wts[7] * 0.15f;
    const float cw4  = wts[9];

    // Wait for the async LDS copy, then read our own lane's slot.
    asm volatile("s_wait_asynccnt 0" ::: "memory");
    const float4 m = ((const float4*)lms)[tid];

#pragma unroll
    for (int b = 0; b < NB; ++b) {
        const float4 sv = ((const float4*)(symtab + (size_t)t[b]   * DIM))[tid];
        const float4 lv = ((const float4*)(lvltab + (size_t)lvl[b] * DIM))[tid];

        vfloat4 o;
        o.x = m.x + cw4 * sv.x + cw2s * lv.x;
        o.y = m.y + cw4 * sv.y + cw2s * lv.y;
        o.z = m.z + cw4 * sv.z + cw2s * lv.z;
        o.w = m.w + cw4 * sv.w + cw2s * lv.w;

        if (opm[b]) {  // uniform across the block (same token for all lanes)
            const float4 pv = ((const float4*)(prectab + (size_t)pid[b] * DIM))[tid];
            o.x += cw1s * pv.x; o.y += cw1s * pv.y;
            o.z += cw1s * pv.z; o.w += cw1s * pv.w;
        }
        vfloat4* dst = (vfloat4*)(out + ((size_t)b * SEQ + s) * DIM) + tid;
        __builtin_nontemporal_store(o, dst);
    }
}

// ---------------------------------------------------------------------------
extern "C" void kernel_launch(void* const* d_in, const int* in_sizes, int n_in,
                              void* d_out, int out_size, void* d_ws, size_t ws_size,
                              hipStream_t stream) {
    const int*   tok     = (const int*)d_in[0];
    const float* symtab  = (const float*)d_in[1];
    const float* lvltab  = (const float*)d_in[2];
    const float* prectab = (const float*)d_in[3];
    const float* sw      = (const float*)d_in[4];
    const float* fa      = (const float*)d_in[5];
    const float* cw      = (const float*)d_in[6];
    float* out = (float*)d_out;

    char* ws = (char*)d_ws;
    float* msad   = (float*)ws;                                          // 16 MB
    int*   levels = (int*)(ws + (size_t)SEQ * DIM * sizeof(float));      // 64 KB
    float* wts    = (float*)(ws + (size_t)SEQ * DIM * sizeof(float)
                                + (size_t)NB * SEQ * sizeof(int));       // 40 B

    softmax2_kernel<<<1, 32, 0, stream>>>(sw, cw, wts);
    level_scan_kernel<<<NB, 256, 0, stream>>>(tok, levels);
    msad_kernel<<<(SEQ * NPAIR) / 256, 256, 0, stream>>>(fa, wts, (float2*)msad);
    combine_kernel<<<SEQ, 256, 0, stream>>>(tok, symtab, lvltab, prectab, wts,
                                            msad, levels, out);
}